// TorsoGATv1_78168404787868
// MI455X (gfx1250) — compile-verified
//
#include <hip/hip_runtime.h>
#include <hip/hip_bf16.h>
#include <math.h>

typedef __attribute__((ext_vector_type(2))) float v2f;
typedef __attribute__((ext_vector_type(8))) float v8f;

#define WG 256

__device__ __forceinline__ int f2ord(float f) {
  int i = __float_as_int(f);
  return i >= 0 ? i : (i ^ 0x7fffffff);
}
__device__ __forceinline__ float ord2f(int i) {
  return __int_as_float(i >= 0 ? i : (i ^ 0x7fffffff));
}

// src/dst for edge e: first E edges from edge_index, then N self loops
__device__ __forceinline__ void edge_nodes(const int* __restrict__ ei, int E, int e,
                                           int& sn, int& dn) {
  if (e < E) { sn = ei[e]; dn = ei[E + e]; }
  else       { sn = e - E; dn = e - E; }
}

__global__ void fill_f32_k(float* p, float v, long n) {
  long i = (long)blockIdx.x * blockDim.x + threadIdx.x;
  if (i < n) p[i] = v;
}
__global__ void fill_i32_k(int* p, int v, long n) {
  long i = (long)blockIdx.x * blockDim.x + threadIdx.x;
  if (i < n) p[i] = v;
}

// ---------------------------------------------------------------------------
// C[M,Nc] = A[M,K] @ B[K,Nc] in fp32 via V_WMMA_F32_16X16X4_F32.
// One wave32 computes a 16 x (16*NT) strip: A fragment reused NT times per
// K-step, NT back-to-back WMMAs. Requires M%16==0, K%4==0, Nc%(16*NT)==0.
// A 16x4 layout : lane L holds M=L%16, K = (L>=16?2:0)+reg  (2 VGPRs)
// B 4x16 layout : lane L holds N=L%16, K = (L>=16?2:0)+reg  (2 VGPRs)
// C/D 16x16     : lane L holds N=L%16, M = reg + 8*(L/16)   (8 VGPRs)
// ---------------------------------------------------------------------------
template <int NT>
__global__ void gemm_wmma_f32_k(const float* __restrict__ A, const float* __restrict__ B,
                                float* __restrict__ C, int M, int K, int Nc) {
  int wave = (int)(((long)blockIdx.x * blockDim.x + threadIdx.x) >> 5);
  int lane = threadIdx.x & 31;
  int groupsN = Nc / (16 * NT);
  int tiles   = (M >> 4) * groupsN;
  if (wave >= tiles) return;                 // wave-uniform: EXEC stays all-1s
  int tm   = (wave / groupsN) << 4;
  int tn0  = (wave % groupsN) * (16 * NT);
  int half = lane >> 4;                      // 0 or 1
  int l16  = lane & 15;

  const float* arow = A + (size_t)(tm + l16) * K;
  v8f acc[NT];
#pragma unroll
  for (int j = 0; j < NT; ++j) acc[j] = (v8f){};

  for (int k0 = 0; k0 < K; k0 += 4) {
    int ka = k0 + half * 2;
    v2f a;
    a.x = arow[ka];
    a.y = arow[ka + 1];
    const float* b0 = B + (size_t)ka * Nc + tn0 + l16;
    const float* b1 = B + (size_t)(ka + 1) * Nc + tn0 + l16;
#pragma unroll
    for (int j = 0; j < NT; ++j) {
      v2f b;
      b.x = b0[16 * j];
      b.y = b1[16 * j];
      acc[j] = __builtin_amdgcn_wmma_f32_16x16x4_f32(
          /*neg_a=*/false, a, /*neg_b=*/false, b,
          /*c_mod=*/(short)0, acc[j], /*reuse_a=*/false, /*reuse_b=*/false);
    }
  }
#pragma unroll
  for (int j = 0; j < NT; ++j)
#pragma unroll
    for (int r = 0; r < 8; ++r)
      C[(size_t)(tm + r + half * 8) * Nc + tn0 + 16 * j + l16] = acc[j][r];
}

// per-node attention logits, one wave32 per (n,h): coalesced channel reads +
// __shfl_xor wave reduction.  s[n,h] = <h[n,h,:], a_s[h,:]>, d likewise.
__global__ void node_logits_k(const float* __restrict__ h, const float* __restrict__ as,
                              const float* __restrict__ ad, float* __restrict__ s,
                              float* __restrict__ d, int N, int H, int C) {
  long wave = ((long)blockIdx.x * blockDim.x + threadIdx.x) >> 5;
  int lane = threadIdx.x & 31;
  if (wave >= (long)N * H) return;
  int n = (int)(wave / H), hh = (int)(wave % H);
  const float* hp = h + (size_t)n * H * C + (size_t)hh * C;
  const float* ap = as + (size_t)hh * C;
  const float* dp = ad + (size_t)hh * C;
  float accs = 0.f, accd = 0.f;
  for (int c = lane; c < C; c += 32) {
    float v = hp[c];
    accs += v * ap[c];
    accd += v * dp[c];
  }
#pragma unroll
  for (int off = 16; off > 0; off >>= 1) {
    accs += __shfl_xor(accs, off, 32);
    accd += __shfl_xor(accd, off, 32);
  }
  if (lane == 0) { s[wave] = accs; d[wave] = accd; }
}

// pass A: e = LeakyReLU(s[src]+d[dst]); store e; segment max over dst
__global__ void edge_max_k(const float* __restrict__ s, const float* __restrict__ d,
                           const int* __restrict__ ei, int E, int N, int H,
                           float* __restrict__ ebuf, int* __restrict__ emax) {
  long t = (long)blockIdx.x * blockDim.x + threadIdx.x;
  long tot = (long)(E + N) * H;
  if (t >= tot) return;
  int e = (int)(t / H), hh = (int)(t % H);
  int sn, dn; edge_nodes(ei, E, e, sn, dn);
  float v = s[(size_t)sn * H + hh] + d[(size_t)dn * H + hh];
  v = v > 0.f ? v : 0.2f * v;
  ebuf[t] = v;
  atomicMax(&emax[(size_t)dn * H + hh], f2ord(v));
}

// pass B: ex = exp(e - max[dst]); store ex; segment sum over dst
__global__ void edge_exp_k(float* __restrict__ ebuf, const int* __restrict__ emax,
                           const int* __restrict__ ei, int E, int N, int H,
                           float* __restrict__ den) {
  long t = (long)blockIdx.x * blockDim.x + threadIdx.x;
  long tot = (long)(E + N) * H;
  if (t >= tot) return;
  int e = (int)(t / H), hh = (int)(t % H);
  int sn, dn; edge_nodes(ei, E, e, sn, dn);
  float m = ord2f(emax[(size_t)dn * H + hh]);
  float ex = expf(ebuf[t] - m);
  ebuf[t] = ex;
  atomicAdd(&den[(size_t)dn * H + hh], ex);
}

// pass C: alpha = ex/(den[dst]+eps); out[dst,h,c] += h[src,h,c] * alpha.
// Thread per (edge, head, channel), channel fastest -> coalesced gathers and
// coalesced global_atomic_add_f32.
__global__ void edge_agg_k(const float* __restrict__ ebuf, const float* __restrict__ den,
                           const int* __restrict__ ei, int E, int N, int H, int C,
                           const float* __restrict__ h, float* __restrict__ out) {
  long t = (long)blockIdx.x * blockDim.x + threadIdx.x;
  long tot = (long)(E + N) * H * C;
  if (t >= tot) return;
  int c  = (int)(t % C);
  long eh = t / C;
  int hh = (int)(eh % H);
  int e  = (int)(eh / H);
  int sn, dn; edge_nodes(ei, E, e, sn, dn);
  float alpha = ebuf[eh] / (den[(size_t)dn * H + hh] + 1e-16f);
  float v = h[(size_t)sn * H * C + (size_t)hh * C + c] * alpha;
  atomicAdd(&out[(size_t)dn * H * C + (size_t)hh * C + c], v);
}

__global__ void bias_relu_k(float* __restrict__ buf, const float* __restrict__ b,
                            long n, int Fmask) {
  long t = (long)blockIdx.x * blockDim.x + threadIdx.x;
  if (t >= n) return;
  float v = buf[t] + b[t & Fmask];
  buf[t] = v > 0.f ? v : 0.f;
}

__global__ void pool_accum_k(const float* __restrict__ h3, const int* __restrict__ batch,
                             float* __restrict__ pooled, float* __restrict__ counts,
                             int N, int C) {
  long t = (long)blockIdx.x * blockDim.x + threadIdx.x;
  if (t >= (long)N * C) return;
  int n = (int)(t / C), c = (int)(t % C);
  int g = batch[n];
  atomicAdd(&pooled[(size_t)g * C + c], h3[t]);
  if (c == 0) atomicAdd(&counts[g], 1.0f);
}

__global__ void final_linear_k(const float* __restrict__ pooled, const float* __restrict__ counts,
                               const float* __restrict__ Wlin, const float* __restrict__ blin,
                               float* __restrict__ out, int G, int Cin, int F) {
  long t = (long)blockIdx.x * blockDim.x + threadIdx.x;
  if (t >= (long)G * F) return;
  int g = (int)(t / F), j = (int)(t % F);
  float inv = 1.0f / fmaxf(counts[g], 1.0f);
  float acc = blin[j];
  const float* pg = pooled + (size_t)g * Cin;
  for (int c = 0; c < Cin; ++c) acc += pg[c] * inv * Wlin[(size_t)c * F + j];
  out[t] = acc;
}

static inline long cdiv(long a, long b) { return (a + b - 1) / b; }

// one full GAT layer: xin[N,K] -> bufB[N,H*C] (post bias+relu). bufA clobbered.
static void run_gat_layer(const float* xin, const float* W, const float* as, const float* ad,
                          const float* b, int K, int H, int C, int N, int E,
                          const int* ei, float* bufA, float* bufB, float* sbuf, float* dbuf,
                          float* ebuf, float* denb, int* emax, hipStream_t stream) {
  const int F = H * C;
  const long E2H = (long)(E + N) * H;
  const int NEG_INF_ORD = (int)0x807FFFFFu;  // f2ord(-inf)

  // GEMM: bufA = xin @ W  (NT=4 strip when F%64==0, else NT=2)
  if (F % 64 == 0) {
    long waves = (long)(N / 16) * (F / 64);
    gemm_wmma_f32_k<4><<<(int)cdiv(waves * 32, 128), 128, 0, stream>>>(xin, W, bufA, N, K, F);
  } else {
    long waves = (long)(N / 16) * (F / 32);
    gemm_wmma_f32_k<2><<<(int)cdiv(waves * 32, 128), 128, 0, stream>>>(xin, W, bufA, N, K, F);
  }
  node_logits_k<<<(int)cdiv((long)N * H * 32, WG), WG, 0, stream>>>(bufA, as, ad, sbuf, dbuf,
                                                                    N, H, C);
  fill_i32_k<<<(int)cdiv((long)N * H, WG), WG, 0, stream>>>(emax, NEG_INF_ORD, (long)N * H);
  fill_f32_k<<<(int)cdiv((long)N * H, WG), WG, 0, stream>>>(denb, 0.f, (long)N * H);
  edge_max_k<<<(int)cdiv(E2H, WG), WG, 0, stream>>>(sbuf, dbuf, ei, E, N, H, ebuf, emax);
  edge_exp_k<<<(int)cdiv(E2H, WG), WG, 0, stream>>>(ebuf, emax, ei, E, N, H, denb);
  fill_f32_k<<<(int)cdiv((long)N * F, WG), WG, 0, stream>>>(bufB, 0.f, (long)N * F);
  edge_agg_k<<<(int)cdiv(E2H * C, WG), WG, 0, stream>>>(ebuf, denb, ei, E, N, H, C, bufA, bufB);
  bias_relu_k<<<(int)cdiv((long)N * F, WG), WG, 0, stream>>>(bufB, b, (long)N * F, F - 1);
}

extern "C" void kernel_launch(void* const* d_in, const int* in_sizes, int n_in,
                              void* d_out, int out_size, void* d_ws, size_t ws_size,
                              hipStream_t stream) {
  // inputs (setup_inputs order)
  const float* x     = (const float*)d_in[0];
  const int*   ei    = (const int*)d_in[1];   // [2,E] flat: row0=src, row1=dst
  const int*   batch = (const int*)d_in[2];
  const float* W1 = (const float*)d_in[3];  const float* as1 = (const float*)d_in[4];
  const float* ad1 = (const float*)d_in[5]; const float* b1  = (const float*)d_in[6];
  const float* W2 = (const float*)d_in[7];  const float* as2 = (const float*)d_in[8];
  const float* ad2 = (const float*)d_in[9]; const float* b2  = (const float*)d_in[10];
  const float* W3 = (const float*)d_in[11]; const float* as3 = (const float*)d_in[12];
  const float* ad3 = (const float*)d_in[13]; const float* b3 = (const float*)d_in[14];
  const float* Wlin = (const float*)d_in[15]; const float* blin = (const float*)d_in[16];
  float* out = (float*)d_out;

  const int N = in_sizes[2];          // 50000 (multiple of 16)
  const int E = in_sizes[1] / 2;      // 800000
  const int G = out_size / 768;       // 256

  // workspace layout (floats)
  float* ws    = (float*)d_ws;
  float* bufA  = ws;                         // N*256  (gemm output / h source)
  float* bufB  = bufA + (size_t)N * 256;     // N*256  (agg target -> layer output)
  float* sbuf  = bufB + (size_t)N * 256;     // N*4
  float* dbuf  = sbuf + (size_t)N * 4;       // N*4
  float* ebuf  = dbuf + (size_t)N * 4;       // (E+N)*4
  float* denb  = ebuf + (size_t)(E + N) * 4; // N*4
  int*   emax  = (int*)(denb + (size_t)N * 4);     // N*4
  float* pooled = (float*)(emax + (size_t)N * 4);  // G*32
  float* counts = pooled + (size_t)G * 32;         // G

  // layer 1: x[N,128] -> h1[N,256] (H=4,C=64), output in bufB
  run_gat_layer(x, W1, as1, ad1, b1, 128, 4, 64, N, E, ei,
                bufA, bufB, sbuf, dbuf, ebuf, denb, emax, stream);
  // layer 2: h1[N,256] -> h2[N,64] (H=1,C=64), output in bufB
  run_gat_layer(bufB, W2, as2, ad2, b2, 256, 1, 64, N, E, ei,
                bufA, bufB, sbuf, dbuf, ebuf, denb, emax, stream);
  // layer 3: h2[N,64] -> h3[N,32] (H=1,C=32), output in bufB
  run_gat_layer(bufB, W3, as3, ad3, b3, 64, 1, 32, N, E, ei,
                bufA, bufB, sbuf, dbuf, ebuf, denb, emax, stream);

  // global mean pool + final linear
  fill_f32_k<<<(int)cdiv((long)G * 32, WG), WG, 0, stream>>>(pooled, 0.f, (long)G * 32);
  fill_f32_k<<<(int)cdiv((long)G, WG), WG, 0, stream>>>(counts, 0.f, (long)G);
  pool_accum_k<<<(int)cdiv((long)N * 32, WG), WG, 0, stream>>>(bufB, batch, pooled, counts, N, 32);
  final_linear_k<<<(int)cdiv((long)G * 768, WG), WG, 0, stream>>>(pooled, counts, Wlin, blin,
                                                                  out, G, 32, 768);
}